// DetectionLoss_79663053406356
// MI455X (gfx1250) — compile-verified
//
#include <hip/hip_runtime.h>
#include <math.h>

// ---- CDNA5 vector types for WMMA ----
typedef __attribute__((ext_vector_type(2))) float v2f;
typedef __attribute__((ext_vector_type(8))) float v8f;

#define NUM_CLASSES 80
#define CCH 85              // 5 + NUM_CLASSES
#define STAL_GAMMA 2.0f

__device__ __forceinline__ float bce_logits(float x, float t) {
    // max(x,0) - x*t + log1p(exp(-|x|))
    return fmaxf(x, 0.0f) - x * t + log1pf(expf(-fabsf(x)));
}

__device__ __forceinline__ float sigmoidf(float x) {
    return 1.0f / (1.0f + expf(-x));
}

// ---------------- zero workspace ----------------
__global__ void dl_zero_kernel(float* __restrict__ p, int n) {
    int i = blockIdx.x * blockDim.x + threadIdx.x;
    if (i < n) p[i] = 0.0f;
}

// ---------------- scatter objectness targets ----------------
__global__ void dl_scatter_obj_kernel(const float* __restrict__ targets, int n,
                                      float* __restrict__ obj0,
                                      float* __restrict__ obj1,
                                      float* __restrict__ obj2) {
    int t = blockIdx.x * blockDim.x + threadIdx.x;
    if (t >= n) return;
    const float* tg = targets + (size_t)t * 6;
    int   b  = (int)tg[0];
    float cx = tg[2], cy = tg[3], bw = tg[4], bh = tg[5];
    float area = fmaxf(bw * bh, 1e-6f);
    int s = (area <= 0.01f) ? 0 : ((area <= 0.03f) ? 1 : 2);
    int w = (s == 0) ? 80 : ((s == 1) ? 40 : 20);
    int h = w;
    int gx = min(max((int)(cx * (float)w), 0), w - 1);
    int gy = min(max((int)(cy * (float)h), 0), h - 1);
    float* m = (s == 0) ? obj0 : ((s == 1) ? obj1 : obj2);
    m[((size_t)b * h + gy) * w + gx] = 1.0f;   // same-value write races are benign
}

// ---------------- per-target box + class loss (WMMA class-BCE reduction) ----------------
// One wave handles 16 targets. 20 iterations of V_WMMA_F32_16X16X4_F32 with
// B = ones accumulate the 80-class BCE row-sums per target into C.
__global__ void dl_target_loss_kernel(const float* __restrict__ p0,
                                      const float* __restrict__ p1,
                                      const float* __restrict__ p2,
                                      const float* __restrict__ targets, int n,
                                      float* __restrict__ accum /*[0]=box,[1]=cls*/) {
    const int lane = threadIdx.x & 31;
    const int wave = (blockIdx.x * blockDim.x + threadIdx.x) >> 5;
    const int m16  = lane & 15;

    int  t     = wave * 16 + m16;
    float maskf = (t < n) ? 1.0f : 0.0f;
    t = (t < n) ? t : (n - 1);               // clamp (no divergence), masked to 0

    const float* tg = targets + (size_t)t * 6;
    int   b   = (int)tg[0];
    int   cls = (int)tg[1];
    float cx = tg[2], cy = tg[3], bw = tg[4], bh = tg[5];

    float area = fmaxf(bw * bh, 1e-6f);
    int s = (area <= 0.01f) ? 0 : ((area <= 0.03f) ? 1 : 2);
    int w = (s == 0) ? 80 : ((s == 1) ? 40 : 20);
    int h = w;
    int hw = h * w;
    const float* pred = (s == 0) ? p0 : ((s == 1) ? p1 : p2);
    int gx = min(max((int)(cx * (float)w), 0), w - 1);
    int gy = min(max((int)(cy * (float)h), 0), h - 1);
    const float* base = pred + (size_t)b * CCH * hw + (size_t)gy * w + gx; // channel 0

    // ---- class BCE row-sum via WMMA f32 16x16x4, B = ones ----
    v8f c = {0.f, 0.f, 0.f, 0.f, 0.f, 0.f, 0.f, 0.f};
    v2f ones; ones.x = 1.0f; ones.y = 1.0f;
    const int chalf = (lane >> 4) << 1;      // 0 for lanes 0-15, 2 for lanes 16-31

    #pragma unroll
    for (int k = 0; k < 20; ++k) {
        int c0 = 4 * k + chalf;
        float x0 = base[(size_t)(5 + c0) * hw];
        float x1 = base[(size_t)(6 + c0) * hw];
        float t0 = (c0     == cls) ? 1.0f : 0.0f;
        float t1 = (c0 + 1 == cls) ? 1.0f : 0.0f;
        v2f a;
        a.x = bce_logits(x0, t0) * maskf;
        a.y = bce_logits(x1, t1) * maskf;
        // (neg_a, A, neg_b, B, c_mod, C, reuse_a, reuse_b)
        c = __builtin_amdgcn_wmma_f32_16x16x4_f32(false, a, false, ones,
                                                  (short)0, c, false, false);
    }
    // Each D column replicates the row sums; lanes 0-15 hold rows 0-7,
    // lanes 16-31 hold rows 8-15 => wave_sum(colsum) == 16 * total.
    float colsum = c[0] + c[1] + c[2] + c[3] + c[4] + c[5] + c[6] + c[7];

    // ---- box loss on lanes 0..15 (target = lane) ----
    float boxc = 0.0f;
    if (lane < 16) {
        float pv0 = base[0];
        float pv1 = base[(size_t)1 * hw];
        float pv2 = base[(size_t)2 * hw];
        float pv3 = base[(size_t)3 * hw];
        float px = (sigmoidf(pv0) + (float)gx) / (float)w;
        float py = (sigmoidf(pv1) + (float)gy) / (float)h;
        float pw = expf(fminf(pv2, 4.0f)) / (float)w;
        float ph = expf(fminf(pv3, 4.0f)) / (float)h;
        float l1 = fabsf(px - cx) + fabsf(py - cy) + fabsf(pw - bw) + fabsf(ph - bh);
        float wgt = 1.0f + STAL_GAMMA * (1.0f - sqrtf(area));
        boxc = l1 * wgt * maskf;             // /4 folded into finalize
    }

    // ---- wave reductions ----
    float vc = colsum;
    float vb = boxc;
    #pragma unroll
    for (int off = 16; off > 0; off >>= 1) {
        vc += __shfl_down(vc, off, 32);
        vb += __shfl_down(vb, off, 32);
    }
    if (lane == 0) {
        atomicAdd(&accum[1], vc * (1.0f / 16.0f));  // /80 folded into finalize
        atomicAdd(&accum[0], vb);
    }
}

// ---------------- objectness BCE over full maps ----------------
__global__ void dl_obj_loss_kernel(const float* __restrict__ p0,
                                   const float* __restrict__ p1,
                                   const float* __restrict__ p2,
                                   const float* __restrict__ obj0,
                                   const float* __restrict__ obj1,
                                   const float* __restrict__ obj2,
                                   int B, float* __restrict__ accum) {
    const int n0 = B * 6400, n1 = B * 1600, n2 = B * 400;
    const int total = n0 + n1 + n2;
    int i = blockIdx.x * blockDim.x + threadIdx.x;

    float contrib = 0.0f;
    if (i < total) {
        const float* pred; const float* obj; int hw; int idx; float inv;
        if (i < n0)            { pred = p0; obj = obj0; hw = 6400; idx = i;          inv = 1.0f / (float)n0; }
        else if (i < n0 + n1)  { pred = p1; obj = obj1; hw = 1600; idx = i - n0;     inv = 1.0f / (float)n1; }
        else                   { pred = p2; obj = obj2; hw = 400;  idx = i - n0 - n1; inv = 1.0f / (float)n2; }
        int b = idx / hw;
        int r = idx - b * hw;
        float x = pred[((size_t)b * CCH + 4) * hw + r];
        contrib = bce_logits(x, obj[idx]) * inv;
    }

    // block reduction: wave shuffle then LDS across waves
    __shared__ float sdata[8];
    int lane = threadIdx.x & 31;
    int wid  = threadIdx.x >> 5;
    #pragma unroll
    for (int off = 16; off > 0; off >>= 1)
        contrib += __shfl_down(contrib, off, 32);
    if (lane == 0) sdata[wid] = contrib;
    __syncthreads();
    if (wid == 0) {
        int nw = blockDim.x >> 5;
        float v = (lane < nw) ? sdata[lane] : 0.0f;
        #pragma unroll
        for (int off = 4; off > 0; off >>= 1)
            v += __shfl_down(v, off, 32);
        if (lane == 0) atomicAdd(&accum[2], v);
    }
}

// ---------------- finalize ----------------
__global__ void dl_finalize_kernel(const float* __restrict__ accum,
                                   float* __restrict__ out, int n) {
    float norm = (float)(n > 0 ? n : 1);
    float box  = accum[0] / (4.0f * norm);
    float clsv = accum[1] / ((float)NUM_CLASSES * norm);
    float obj  = accum[2];
    out[0] = box + obj + clsv;
    out[1] = box;
    out[2] = obj;
    out[3] = clsv;
}

extern "C" void kernel_launch(void* const* d_in, const int* in_sizes, int n_in,
                              void* d_out, int out_size, void* d_ws, size_t ws_size,
                              hipStream_t stream) {
    const float* p0      = (const float*)d_in[0];   // (B,85,80,80)
    const float* p1      = (const float*)d_in[1];   // (B,85,40,40)
    const float* p2      = (const float*)d_in[2];   // (B,85,20,20)
    const float* targets = (const float*)d_in[3];   // (N,6)

    int B = in_sizes[0] / (CCH * 6400);
    int n = in_sizes[3] / 6;

    float* ws    = (float*)d_ws;
    float* accum = ws;                         // [0]=box, [1]=cls, [2]=obj
    float* obj0  = ws + 16;
    float* obj1  = obj0 + (size_t)B * 6400;
    float* obj2  = obj1 + (size_t)B * 1600;

    int zn = 16 + B * (6400 + 1600 + 400);
    dl_zero_kernel<<<(zn + 255) / 256, 256, 0, stream>>>(ws, zn);

    dl_scatter_obj_kernel<<<(n + 255) / 256, 256, 0, stream>>>(targets, n, obj0, obj1, obj2);

    int waves   = (n + 15) / 16;
    int threads = waves * 32;
    dl_target_loss_kernel<<<(threads + 255) / 256, 256, 0, stream>>>(p0, p1, p2, targets, n, accum);

    int tot = B * (6400 + 1600 + 400);
    dl_obj_loss_kernel<<<(tot + 255) / 256, 256, 0, stream>>>(p0, p1, p2, obj0, obj1, obj2, B, accum);

    dl_finalize_kernel<<<1, 1, 0, stream>>>(accum, (float*)d_out, n);
}